// ProteinTraversalSpatial_66434554134872
// MI455X (gfx1250) — compile-verified
//
#include <hip/hip_runtime.h>

// Problem constants (match reference)
#define NB   4
#define NN   4096
#define KNB  30
#define BIGV 1.0e9f
#define CONSUMED 3.0e9f

typedef __attribute__((ext_vector_type(2))) float v2f;
typedef __attribute__((ext_vector_type(8))) float v8f;

// ---------------------------------------------------------------------------
// Kernel 0: extract atom-1 coords, precompute squared norms.
// coords4[g] = (x, y, z, |x|^2)
// ---------------------------------------------------------------------------
__global__ void prep_kernel(const float* __restrict__ X, float4* __restrict__ coords4) {
    int g = blockIdx.x * blockDim.x + threadIdx.x;
    if (g >= NB * NN) return;
    const float* p = X + ((size_t)g * 14 + 1) * 3;
    float x = p[0], y = p[1], z = p[2];
    coords4[g] = make_float4(x, y, z, x * x + y * y + z * z);
}

// ---------------------------------------------------------------------------
// Kernel 1: fused KNN. One block = 16 rows of one batch.
//   Stage 1: 16 waves cooperatively fill a 16 x 4096 d2 panel in LDS using
//            V_WMMA_F32_16X16X4_F32 (A = 16x4 coords, B = 4x16 coords^T).
//   Stage 2: wave r selects top-30 smallest d2 of panel row r.
// ---------------------------------------------------------------------------
__global__ __launch_bounds__(512) void knn_kernel(const float4* __restrict__ coords4,
                                                  const int* __restrict__ C,
                                                  int* __restrict__ eidx,
                                                  int* __restrict__ ecnt) {
    __shared__ float panel[16][NN];   // 256 KB (CDNA5 WGP: up to 320 KB)
    __shared__ float sqi_s[16];
    __shared__ int   maski_s[16];

    const int b     = blockIdx.x >> 8;          // 256 blocks per batch
    const int ibase = (blockIdx.x & 255) << 4;  // 16-row panel
    const int tid   = threadIdx.x;
    const int lane  = tid & 31;
    const int wave  = tid >> 5;

    if (tid < 16) {
        float4 ci = coords4[b * NN + ibase + tid];
        sqi_s[tid]   = ci.w;
        maski_s[tid] = (C[b * NN + ibase + tid] > 0) ? 1 : 0;
    }
    __syncthreads();

    const int q    = lane & 15;   // row-within-tile this lane supplies
    const int half = lane >> 4;   // K half: 0 -> K{0,1}, 1 -> K{2,3(pad)}

    // A matrix (16x4, f32): lanes 0-15 hold K0/K1, lanes 16-31 hold K2/K3.
    float4 ca = coords4[b * NN + ibase + q];
    v2f Areg;
    Areg.x = half ? ca.z : ca.x;
    Areg.y = half ? 0.0f : ca.y;

    // ---- Stage 1: fill panel with masked squared distances -----------------
    for (int jt = wave; jt < NN / 16; jt += 16) {
        const int jb = jt << 4;
        float4 cb = coords4[b * NN + jb + q];
        const int maskj = (C[b * NN + jb + q] > 0) ? 1 : 0;

        v2f Breg;                       // B matrix (4x16): same per-lane gather
        Breg.x = half ? cb.z : cb.x;
        Breg.y = half ? 0.0f : cb.y;

        v8f acc = {0.f, 0.f, 0.f, 0.f, 0.f, 0.f, 0.f, 0.f};
        acc = __builtin_amdgcn_wmma_f32_16x16x4_f32(
            /*neg_a=*/false, Areg, /*neg_b=*/false, Breg,
            /*c_mod=*/(short)0, acc, /*reuse_a=*/false, /*reuse_b=*/false);

        const float sqj = cb.w;         // lane's column n == q
#pragma unroll
        for (int v = 0; v < 8; ++v) {
            const int m = v + (half << 3);              // D layout: M = v + 8*(lane>=16)
            const float d2 = sqi_s[m] + sqj - 2.0f * acc[v];
            const int valid = maski_s[m] & maskj;
            panel[m][jb + q] = valid ? d2 : BIGV;
        }
    }
    __syncthreads();

    // ---- Stage 2: top-30 selection, wave r owns panel row r ----------------
    float* row = panel[wave];

    float bestv = CONSUMED;
    int   bestj = 0x7fffffff;
    for (int j = lane; j < NN; j += 32) {     // ascending scan => lowest idx on ties
        float v = row[j];
        if (v < bestv) { bestv = v; bestj = j; }
    }

    const int gi = b * NN + ibase + wave;
    const int mi = maski_s[wave];
    int cnt = 0;

    for (int k = 0; k < KNB; ++k) {
        float rv = bestv;
        int   rj = bestj;
#pragma unroll
        for (int off = 16; off > 0; off >>= 1) {       // lexicographic wave-min
            float ov = __shfl_xor(rv, off, 32);
            int   oj = __shfl_xor(rj, off, 32);
            if (ov < rv || (ov == rv && oj < rj)) { rv = ov; rj = oj; }
        }
        if (lane == 0) eidx[(size_t)gi * KNB + k] = rj;
        cnt += (mi && rv < 0.5f * BIGV) ? 1 : 0;       // valid neighbor pair

        if (bestj == rj) {                              // unique owner pops + rescans
            row[rj] = CONSUMED;
            bestv = CONSUMED;
            bestj = 0x7fffffff;
            for (int j = lane; j < NN; j += 32) {
                float v = row[j];
                if (v < bestv) { bestv = v; bestj = j; }
            }
        }
    }
    if (lane == 0) ecnt[gi] = cnt;
}

// ---------------------------------------------------------------------------
// Kernel 2: one Jacobi smoothing step (launched 5x, ping-pong).
// Valid neighbors are the first ecnt entries (pops are ascending in d2, so
// all finite/valid pairs precede BIG/masked ones). denom = cnt + 1e-5.
// ---------------------------------------------------------------------------
__global__ void smooth_kernel(const float* __restrict__ zin, float* __restrict__ zout,
                              const int* __restrict__ eidx, const int* __restrict__ ecnt) {
    int g = blockIdx.x * blockDim.x + threadIdx.x;
    if (g >= NB * NN) return;
    const int base = g & ~(NN - 1);
    const int c = ecnt[g];
    const int* e = eidx + (size_t)g * KNB;
    float s = 0.0f;
    for (int k = 0; k < c; ++k) s += zin[base + e[k]];
    const float avg = s / ((float)c + 1e-5f);
    zout[g] = 0.9f * avg + 0.1f * zin[g];
}

// ---------------------------------------------------------------------------
// Kernel 3: stable argsort via counting rank + scatter (ranks are a perfect
// permutation, so d_out is fully written).
// ---------------------------------------------------------------------------
__global__ void rank_kernel(const float* __restrict__ z, int* __restrict__ out) {
    __shared__ float tile[256];
    const int b = blockIdx.x >> 4;                      // 16 blocks per row
    const int i = ((blockIdx.x & 15) << 8) + threadIdx.x;
    const float* zr = z + b * NN;
    const float zi = zr[i];
    int rank = 0;
    for (int t = 0; t < NN / 256; ++t) {
        tile[threadIdx.x] = zr[t * 256 + threadIdx.x];
        __syncthreads();
#pragma unroll 8
        for (int jj = 0; jj < 256; ++jj) {
            const int j = t * 256 + jj;
            const float zj = tile[jj];
            rank += (zj < zi || (zj == zi && j < i)) ? 1 : 0;
        }
        __syncthreads();
    }
    out[b * NN + rank] = i;
}

// ---------------------------------------------------------------------------
extern "C" void kernel_launch(void* const* d_in, const int* in_sizes, int n_in,
                              void* d_out, int out_size, void* d_ws, size_t ws_size,
                              hipStream_t stream) {
    const float* X = (const float*)d_in[0];   // (4,4096,14,3) f32
    const int*   C = (const int*)d_in[1];     // (4,4096) i32
    const float* z = (const float*)d_in[2];   // (4,4096) f32

    // Workspace layout (~2.4 MB)
    char* ws = (char*)d_ws;
    float4* coords4 = (float4*)ws;            ws += (size_t)NB * NN * sizeof(float4);
    float*  z0      = (float*)ws;             ws += (size_t)NB * NN * sizeof(float);
    float*  z1      = (float*)ws;             ws += (size_t)NB * NN * sizeof(float);
    int*    ecnt    = (int*)ws;               ws += (size_t)NB * NN * sizeof(int);
    int*    eidx    = (int*)ws;

    prep_kernel<<<(NB * NN) / 256, 256, 0, stream>>>(X, coords4);
    knn_kernel<<<NB * (NN / 16), 512, 0, stream>>>(coords4, C, eidx, ecnt);

    smooth_kernel<<<(NB * NN) / 256, 256, 0, stream>>>(z,  z0, eidx, ecnt);
    smooth_kernel<<<(NB * NN) / 256, 256, 0, stream>>>(z0, z1, eidx, ecnt);
    smooth_kernel<<<(NB * NN) / 256, 256, 0, stream>>>(z1, z0, eidx, ecnt);
    smooth_kernel<<<(NB * NN) / 256, 256, 0, stream>>>(z0, z1, eidx, ecnt);
    smooth_kernel<<<(NB * NN) / 256, 256, 0, stream>>>(z1, z0, eidx, ecnt);

    rank_kernel<<<NB * (NN / 256), 256, 0, stream>>>(z0, (int*)d_out);
}